// PatchLevelPruner_2147483648649
// MI455X (gfx1250) — compile-verified
//
#include <hip/hip_runtime.h>
#include <math.h>

typedef __attribute__((ext_vector_type(2))) float v2f;
typedef __attribute__((ext_vector_type(8))) float v8f;
typedef __attribute__((ext_vector_type(4))) float f4;

#define C_DIM   256
#define HID     64
#define NTOK    16384   // tokens per batch (128*128)
#define BATCH   8
#define PPB     1024    // patches per batch (32*32)
#define KKEEP   768
#define OUTROWS 12288   // 24*4*32*4
#define LDSTR   258     // LDS row stride for transposed w1 (even + conflict-free)

// ---------------------------------------------------------------------------
// Kernel 1: per-token importance via WMMA fp32 GEMM [16tok x 256] x [256 x 64]
// h = gelu(x@w1 + b1);  tokenScore = sigmoid(h@w2 + b2)
// ---------------------------------------------------------------------------
__global__ __launch_bounds__(256) void mlp_score_kernel(
    const float* __restrict__ tokens,
    const float* __restrict__ w1,   // [256,64] row-major
    const float* __restrict__ b1,   // [64]
    const float* __restrict__ w2,   // [64]
    const float* __restrict__ b2,   // [1]
    float* __restrict__ tokenScore) // [B*NTOK]
{
    __shared__ float w1t[HID * LDSTR];   // w1 transposed: w1t[d*LDSTR + k]

    const int tid = threadIdx.x;
    for (int idx = tid; idx < C_DIM * HID; idx += 256) {
        int k = idx >> 6;        // 0..255
        int d = idx & 63;        // 0..63
        w1t[d * LDSTR + k] = w1[idx];
    }
    __syncthreads();

    const int lane = tid & 31;
    const int wave = tid >> 5;
    const int l15  = lane & 15;
    const int hi   = lane >> 4;          // 0: K=0,1  1: K=2,3 (A layout halves)
    const float b2v = b2[0];

    float b1v[4], w2v[4];
#pragma unroll
    for (int ct = 0; ct < 4; ++ct) {
        b1v[ct] = b1[ct * 16 + l15];
        w2v[ct] = w2[ct * 16 + l15];
    }

    const int waveGlobal = blockIdx.x * 8 + wave;   // 2048 waves, 4 tiles each
    for (int t = 0; t < 4; ++t) {
        const long tileBase = ((long)waveGlobal * 4 + t) * 16;
        const float* aRow = tokens + (tileBase + (long)l15) * C_DIM + hi * 2;

        v8f acc0 = {}, acc1 = {}, acc2 = {}, acc3 = {};
        for (int k0 = 0; k0 < C_DIM; k0 += 4) {
            v2f a = *(const v2f*)(aRow + k0);             // A[row][k0+2*hi .. +1]
            const int kk = k0 + hi * 2;
            v2f bv0 = *(v2f*)&w1t[( 0 + l15) * LDSTR + kk];
            v2f bv1 = *(v2f*)&w1t[(16 + l15) * LDSTR + kk];
            v2f bv2 = *(v2f*)&w1t[(32 + l15) * LDSTR + kk];
            v2f bv3 = *(v2f*)&w1t[(48 + l15) * LDSTR + kk];
            acc0 = __builtin_amdgcn_wmma_f32_16x16x4_f32(false, a, false, bv0, (short)0, acc0, false, false);
            acc1 = __builtin_amdgcn_wmma_f32_16x16x4_f32(false, a, false, bv1, (short)0, acc1, false, false);
            acc2 = __builtin_amdgcn_wmma_f32_16x16x4_f32(false, a, false, bv2, (short)0, acc2, false, false);
            acc3 = __builtin_amdgcn_wmma_f32_16x16x4_f32(false, a, false, bv3, (short)0, acc3, false, false);
        }

        // epilogue: gelu(h)·w2, reduce across the 16 hidden columns per lane-half
        float partial[8];
#pragma unroll
        for (int r = 0; r < 8; ++r) partial[r] = 0.f;

#pragma unroll
        for (int r = 0; r < 8; ++r) {
            float x, g;
            x = acc0[r] + b1v[0]; g = 0.5f * x * (1.f + erff(x * 0.70710678f)); partial[r] += g * w2v[0];
            x = acc1[r] + b1v[1]; g = 0.5f * x * (1.f + erff(x * 0.70710678f)); partial[r] += g * w2v[1];
            x = acc2[r] + b1v[2]; g = 0.5f * x * (1.f + erff(x * 0.70710678f)); partial[r] += g * w2v[2];
            x = acc3[r] + b1v[3]; g = 0.5f * x * (1.f + erff(x * 0.70710678f)); partial[r] += g * w2v[3];
        }

#pragma unroll
        for (int r = 0; r < 8; ++r) {
            float v = partial[r];
            v += __shfl_xor(v, 1, 32);
            v += __shfl_xor(v, 2, 32);
            v += __shfl_xor(v, 4, 32);
            v += __shfl_xor(v, 8, 32);           // sum within each 16-lane half
            float s = 1.f / (1.f + expf(-(v + b2v)));
            if (l15 == 0) tokenScore[tileBase + hi * 8 + r] = s;  // rows r / r+8
        }
    }
}

// ---------------------------------------------------------------------------
// Kernel 2: patch score = mean of its 4x4 spatial block of token scores
// ---------------------------------------------------------------------------
__global__ __launch_bounds__(256) void patch_score_kernel(
    const float* __restrict__ tokenScore, float* __restrict__ patchScore)
{
    int gid = blockIdx.x * 256 + threadIdx.x;
    if (gid >= BATCH * PPB) return;
    int b = gid >> 10, p = gid & 1023;
    int py = p >> 5, px = p & 31;
    const float* base = tokenScore + (long)b * NTOK;
    float s = 0.f;
#pragma unroll
    for (int r = 0; r < 4; ++r)
#pragma unroll
        for (int c = 0; c < 4; ++c)
            s += base[(py * 4 + r) * 128 + px * 4 + c];
    patchScore[gid] = s * (1.f / 16.f);
}

// ---------------------------------------------------------------------------
// Kernel 3: top-K by rank counting (jax top_k tie semantics: lower index wins)
// then ascending-order compaction of kept patch indices.
// ---------------------------------------------------------------------------
__global__ __launch_bounds__(1024) void select_kernel(
    const float* __restrict__ patchScore, int* __restrict__ map)
{
    __shared__ float s[PPB];
    __shared__ int keep[PPB];
    const int b = blockIdx.x;
    const int i = threadIdx.x;
    s[i] = patchScore[b * PPB + i];
    __syncthreads();
    const float si = s[i];
    int rank = 0;
    for (int j = 0; j < PPB; ++j) {
        float sj = s[j];
        rank += (sj > si) || (sj == si && j < i);
    }
    keep[i] = (rank < KKEEP) ? 1 : 0;
    __syncthreads();
    if (keep[i]) {
        int pos = 0;
        for (int j = 0; j < i; ++j) pos += keep[j];
        map[b * KKEEP + pos] = i;
    }
}

// ---------------------------------------------------------------------------
// Kernel 4: gather kept patches and reassemble into [B, 24*4*32*4, C]
// one block per (b, kept-slot); float4 row copies.
// ---------------------------------------------------------------------------
__global__ __launch_bounds__(256) void gather_kernel(
    const float* __restrict__ tokens, const int* __restrict__ map,
    float* __restrict__ out)
{
    const int blk = blockIdx.x;        // b*KKEEP + k
    const int b = blk / KKEEP;
    const int k = blk - b * KKEEP;
    const int p = map[blk];
    const int py = p >> 5, px = p & 31;
    const int i  = k >> 5, j  = k & 31;
    const int t    = threadIdx.x >> 4;   // 0..15 patch-local token
    const int part = threadIdx.x & 15;   // 16-float chunk within the channel row
    const int r = t >> 2, c = t & 3;
    const long srcRow = (long)b * NTOK    + (py * 4 + r) * 128 + px * 4 + c;
    const long dstRow = (long)b * OUTROWS + ((i * 4 + r) * 32 + j) * 4 + c;
    const f4* src = (const f4*)(tokens + srcRow * C_DIM) + part * 4;
    f4*       dst = (f4*)      (out    + dstRow * C_DIM) + part * 4;
#pragma unroll
    for (int u = 0; u < 4; ++u) dst[u] = src[u];
}

// ---------------------------------------------------------------------------
extern "C" void kernel_launch(void* const* d_in, const int* in_sizes, int n_in,
                              void* d_out, int out_size, void* d_ws, size_t ws_size,
                              hipStream_t stream) {
    const float* tokens = (const float*)d_in[0];  // [8,16384,256]
    const float* w1     = (const float*)d_in[1];  // [256,64]
    const float* b1     = (const float*)d_in[2];  // [64]
    const float* w2     = (const float*)d_in[3];  // [64,1]
    const float* b2     = (const float*)d_in[4];  // [1]
    // d_in[5]=H, d_in[6]=W are fixed (128,128) — baked into the kernels.
    float* out = (float*)d_out;

    char* ws = (char*)d_ws;
    float* tokenScore = (float*)ws;                                    // 8*16384 f32
    float* patchScore = (float*)(ws + (size_t)BATCH * NTOK * 4);       // 8192 f32
    int*   map        = (int*)  (ws + (size_t)BATCH * NTOK * 4
                                    + (size_t)BATCH * PPB * 4);        // 6144 i32

    // 8192 token tiles / (8 waves * 4 tiles per wave) = 256 blocks
    mlp_score_kernel<<<256, 256, 0, stream>>>(tokens, w1, b1, w2, b2, tokenScore);
    patch_score_kernel<<<(BATCH * PPB + 255) / 256, 256, 0, stream>>>(tokenScore, patchScore);
    select_kernel<<<BATCH, 1024, 0, stream>>>(patchScore, map);
    gather_kernel<<<BATCH * KKEEP, 256, 0, stream>>>(tokens, map, out);
}